// EnergyDipolesMACE_43293270343922
// MI455X (gfx1250) — compile-verified
//
#include <hip/hip_runtime.h>
#include <hip/hip_bf16.h>
#include <math.h>

typedef float v2f __attribute__((ext_vector_type(2)));
typedef float v8f __attribute__((ext_vector_type(8)));

#define C_CH 64
#define R_MAX_F 5.0f

__device__ __forceinline__ float siluf(float x) { return x / (1.0f + __expf(-x)); }

// ---------------------------------------------------------------------------
// Strided GEMM on fp32 WMMA 16x16x4:  Y(row,col) (+)= sum_k op(X(row,k))*W(k,col)
// All strides are compile-time so every global access is base + immediate.
// X element (r,k) at X[m*XMS + r*XRS + k*XKS]; W element (k,col) at
// W[k*WKS + col*WCS] (swapped strides give free transpose); Y at
// Y[m*YMS + r*YRS + col*YCS].  PERL: select W slice by SH block l(m), m=blockIdx.y.
// W staged in LDS, pair-interleaved so each B fragment is one aligned v2f load.
// One wave computes a 16-row x D-col tile.  Requires nRows % 16 == 0.
// ---------------------------------------------------------------------------
template<int K, int D, int PERL, int ACC, int SILU,
         int XRS, int XKS, int XMS, int WKS, int WCS, int YRS, int YCS, int YMS>
__global__ void wmma_gemm(const float* __restrict__ X, const float* __restrict__ W,
                          float* __restrict__ Y, int nRows)
{
    constexpr int NCT = (D + 15) / 16;
    __shared__ float lds_w[(K / 2) * 128];

    const int lane = threadIdx.x & 31;
    const int wave = threadIdx.x >> 5;
    const int rowTile = blockIdx.x * (blockDim.x >> 5) + wave;
    const int row0 = rowTile * 16;
    const int m = blockIdx.y;
    if (PERL) {
        const int l = (m == 0) ? 0 : ((m < 4) ? 1 : 2);
        W += l * C_CH * C_CH;
    }
    // stage W into LDS: lds_w[(k>>1)*128 + col*2 + (k&1)]
    for (int j = threadIdx.x; j < (K / 2) * 128; j += blockDim.x) {
        const int pairrow = j >> 7;
        const int rem = j & 127;
        const int col = rem >> 1;
        const int kin = rem & 1;
        const int k = pairrow * 2 + kin;
        lds_w[j] = (col < D) ? W[k * WKS + col * WCS] : 0.0f;
    }
    __syncthreads();
    if (row0 >= nRows) return;

    const int half = lane >> 4;   // second half-wave: K+2 (A/B), M+8 (C/D)
    const int lid  = lane & 15;

    v8f zero8 = {0.f,0.f,0.f,0.f,0.f,0.f,0.f,0.f};
    v8f acc[NCT];
#pragma unroll
    for (int ct = 0; ct < NCT; ++ct) acc[ct] = zero8;

    // one 64-bit base per lane; all element accesses use immediate offsets
    const float* xrow = X + m * XMS + (row0 + lid) * XRS + half * 2 * XKS;

#pragma unroll
    for (int k0 = 0; k0 < K; k0 += 4) {
        float a0 = xrow[k0 * XKS];
        float a1 = xrow[k0 * XKS + XKS];
        if (SILU) { a0 = siluf(a0); a1 = siluf(a1); }
        v2f a; a.x = a0; a.y = a1;
        const int karow = (k0 >> 1) + half;   // (ka>>1) with ka = k0 + half*2
#pragma unroll
        for (int ct = 0; ct < NCT; ++ct) {
            v2f b = *(const v2f*)&lds_w[karow * 128 + (ct * 16 + lid) * 2];
            acc[ct] = __builtin_amdgcn_wmma_f32_16x16x4_f32(
                false, a, false, b, (short)0, acc[ct], false, false);
        }
    }
    float* ybase = Y + m * YMS + (row0 + half * 8) * YRS + lid * YCS;
#pragma unroll
    for (int ct = 0; ct < NCT; ++ct) {
        if (D < 16 && lid >= D) continue;
        float* yp = ybase + ct * 16 * YCS;
#pragma unroll
        for (int r = 0; r < 8; ++r) {
            if (ACC) yp[r * YRS] += acc[ct][r]; else yp[r * YRS] = acc[ct][r];
        }
    }
}

// --------------------------- pointwise / edge kernels ----------------------

__global__ void edge_geom_k(const float* __restrict__ pos, const float* __restrict__ shifts,
                            const int* __restrict__ snd, const int* __restrict__ rcv,
                            float* __restrict__ vec_o, float* __restrict__ len_o,
                            float* __restrict__ sh_o, float* __restrict__ ef_o, int E)
{
    int e = blockIdx.x * blockDim.x + threadIdx.x;
    if (e >= E) return;
    int s = snd[e], r = rcv[e];
    float vx = pos[s*3+0] - pos[r*3+0] + shifts[e*3+0];
    float vy = pos[s*3+1] - pos[r*3+1] + shifts[e*3+1];
    float vz = pos[s*3+2] - pos[r*3+2] + shifts[e*3+2];
    float l2 = vx*vx + vy*vy + vz*vz + 1e-12f;
    float len = sqrtf(l2);
    vec_o[e*3+0]=vx; vec_o[e*3+1]=vy; vec_o[e*3+2]=vz; len_o[e]=len;
    float inv = 1.0f/len;
    float x = vx*inv, y = vy*inv, z = vz*inv;
    const float s3 = 1.7320508075688772f, s5 = 2.2360679774997896f, s15 = 3.8729833462074170f;
    float sh[9];
    sh[0]=1.0f; sh[1]=s3*x; sh[2]=s3*y; sh[3]=s3*z;
    sh[4]=s15*x*y; sh[5]=s15*y*z; sh[6]=0.5f*s5*(3.0f*z*z-1.0f);
    sh[7]=s15*x*z; sh[8]=0.5f*s15*(x*x-y*y);
#pragma unroll
    for (int m = 0; m < 9; ++m) sh_o[(size_t)e*9+m] = sh[m];
    float u = len * (1.0f/R_MAX_F);
    float fc = 0.0f;
    if (u < 1.0f) {
        float u2=u*u, u5=u2*u2*u;
        fc = 1.0f - 21.0f*u5 + 35.0f*u5*u - 15.0f*u5*u2;
    }
    float A = sqrtf(2.0f/R_MAX_F);
    float rinv = 1.0f/(len + 1e-9f);
    const float PI_F = 3.14159265358979f;
#pragma unroll
    for (int b = 0; b < 8; ++b) {
        float np = (float)(b+1)*PI_F;
        ef_o[(size_t)e*8+b] = A * sinf(np*u) * rinv * fc;
    }
}

__global__ void node_init_k(const float* __restrict__ attrs, const float* __restrict__ W_emb,
                            const float* __restrict__ ae, const int* __restrict__ batch,
                            float* __restrict__ nf, float* __restrict__ contrib, int N, int Z)
{
    int idx = blockIdx.x * blockDim.x + threadIdx.x;
    if (idx >= N*64) return;
    int n = idx >> 6, c = idx & 63;
    float acc = 0.f;
    for (int z = 0; z < Z; ++z) acc += attrs[(size_t)n*Z+z] * W_emb[z*64+c];
    float* p = nf + (size_t)n*576 + c*9;
    p[0] = acc;
#pragma unroll
    for (int m = 1; m < 9; ++m) p[m] = 0.f;
    if (c == 0) {
        float e0 = 0.f;
        for (int z = 0; z < Z; ++z) e0 += attrs[(size_t)n*Z+z] * ae[z];
        atomicAdd(&contrib[batch[n]*3 + 0], e0);
    }
}

__global__ void save_ch0_k(const float* __restrict__ nf, float* __restrict__ nf0, int N)
{
    int idx = blockIdx.x * blockDim.x + threadIdx.x;
    if (idx >= N*64) return;
    nf0[idx] = nf[(size_t)(idx>>6)*576 + (size_t)(idx&63)*9];
}

__global__ void edge_scatter_k(const float* __restrict__ w, const float* __restrict__ nf0,
                               const float* __restrict__ sh, const int* __restrict__ snd,
                               const int* __restrict__ rcv, float* __restrict__ agg, int E)
{
    long idx = (long)blockIdx.x * blockDim.x + threadIdx.x;
    if (idx >= (long)E*64) return;
    int e = (int)(idx >> 6), c = (int)(idx & 63);
    float wh = w[idx] * nf0[(size_t)snd[e]*64 + c] * 0.0625f;  // /AVG_NEIGH
    const float* shp = sh + (size_t)e*9;
    float* ap = agg + (size_t)rcv[e]*576 + c*9;
#pragma unroll
    for (int m = 0; m < 9; ++m) atomicAdd(&ap[m], wh * shp[m]);
}

__global__ void node_poly_fwd_k(const float* __restrict__ msg, const float* __restrict__ wpoly,
                                float* __restrict__ mp, int N)
{
    int idx = blockIdx.x * blockDim.x + threadIdx.x;
    if (idx >= N*64) return;
    int n = idx >> 6, c = idx & 63;
    const float* mrow = msg + (size_t)n*576 + c*9;
    const float* wp = wpoly + c*3;
    float s0 = mrow[0];
    float poly = wp[0] + wp[1]*s0 + wp[2]*s0*s0;
    float* orow = mp + (size_t)n*576 + c*9;
#pragma unroll
    for (int m = 0; m < 9; ++m) orow[m] = mrow[m] * poly;
}

__global__ void energy_dipole_k(const float* __restrict__ nf, const float* __restrict__ we,
                                const float* __restrict__ wd, const int* __restrict__ batch,
                                float* __restrict__ contrib, float* __restrict__ ad,
                                int N, int tcol)
{
    int n = blockIdx.x * blockDim.x + threadIdx.x;
    if (n >= N) return;
    const float* p = nf + (size_t)n*576;
    float en = 0.f, d0 = 0.f, d1 = 0.f, d2 = 0.f;
    for (int c = 0; c < 64; ++c) {
        float w1 = we[c], w2 = wd[c];
        en += p[c*9+0]*w1;
        d0 += p[c*9+1]*w2; d1 += p[c*9+2]*w2; d2 += p[c*9+3]*w2;
    }
    atomicAdd(&contrib[batch[n]*3 + tcol], en);
    ad[n*3+0] += d0; ad[n*3+1] += d1; ad[n*3+2] += d2;
}

__global__ void finalize_dipole_k(const float* __restrict__ ad, const float* __restrict__ charges,
                                  const float* __restrict__ pos, const int* __restrict__ batch,
                                  float* __restrict__ td, int N)
{
    int n = blockIdx.x * blockDim.x + threadIdx.x;
    if (n >= N) return;
    int g = batch[n];
    float q = charges[n];
#pragma unroll
    for (int j = 0; j < 3; ++j) atomicAdd(&td[g*3+j], ad[n*3+j] + q*pos[n*3+j]);
}

__global__ void total_energy_k(const float* __restrict__ contrib, float* __restrict__ te, int G)
{
    int g = blockIdx.x * blockDim.x + threadIdx.x;
    if (g >= G) return;
    te[g] = contrib[g*3+0] + contrib[g*3+1] + contrib[g*3+2];
}

// --------------------------------- backward --------------------------------

__global__ void add_we_k(float* __restrict__ gnf, const float* __restrict__ we, int N)
{
    int idx = blockIdx.x * blockDim.x + threadIdx.x;
    if (idx >= N*64) return;
    gnf[(size_t)(idx>>6)*576 + (size_t)(idx&63)*9] += we[idx & 63];
}

__global__ void node_poly_bwd_k(float* __restrict__ gA, const float* __restrict__ msg,
                                const float* __restrict__ wpoly, int N)
{
    int idx = blockIdx.x * blockDim.x + threadIdx.x;
    if (idx >= N*64) return;
    int n = idx >> 6, c = idx & 63;
    const float* mrow = msg + (size_t)n*576 + c*9;
    const float* wp = wpoly + c*3;
    float s0 = mrow[0];
    float poly = wp[0] + wp[1]*s0 + wp[2]*s0*s0;
    float* grow = gA + (size_t)n*576 + c*9;
    float gm[9]; float gp = 0.f;
#pragma unroll
    for (int m = 0; m < 9; ++m) { gm[m] = grow[m]; gp += gm[m]*mrow[m]; }
#pragma unroll
    for (int m = 0; m < 9; ++m) grow[m] = gm[m]*poly;
    grow[0] += gp * (wp[1] + 2.0f*wp[2]*s0);
}

__global__ void edge_bwd_k(const float* __restrict__ g_agg, const float* __restrict__ w,
                           const float* __restrict__ nf0, const float* __restrict__ sh,
                           const int* __restrict__ snd, const int* __restrict__ rcv,
                           float* __restrict__ g_w, float* __restrict__ g_sh,
                           float* __restrict__ g_nf_in, int E)
{
    int e = blockIdx.x * blockDim.x + threadIdx.x;
    if (e >= E) return;
    int s = snd[e], r = rcv[e];
    const float* shp = sh + (size_t)e*9;
    const float* gap = g_agg + (size_t)r*576;
    float gsh[9] = {0,0,0,0,0,0,0,0,0};
    for (int c = 0; c < 64; ++c) {
        float wv = w[(size_t)e*64 + c];
        float hs = nf0[(size_t)s*64 + c];
        float wh = wv*hs*0.0625f;
        float gwh = 0.f;
#pragma unroll
        for (int m = 0; m < 9; ++m) {
            float ga = gap[c*9+m];
            gwh += ga*shp[m];
            gsh[m] += ga*wh;
        }
        gwh *= 0.0625f;
        g_w[(size_t)e*64 + c] = gwh*hs;
        atomicAdd(&g_nf_in[(size_t)s*576 + c*9], gwh*wv);
    }
#pragma unroll
    for (int m = 0; m < 9; ++m) g_sh[(size_t)e*9+m] += gsh[m];
}

__global__ void silu_bwd_k(float* __restrict__ gh, const float* __restrict__ pre, long n)
{
    long i = (long)blockIdx.x * blockDim.x + threadIdx.x;
    if (i >= n) return;
    float x = pre[i];
    float sg = 1.0f / (1.0f + __expf(-x));
    gh[i] *= sg * (1.0f + x*(1.0f - sg));
}

__global__ void geom_bwd_k(const float* __restrict__ g_sh, const float* __restrict__ g_ef,
                           const float* __restrict__ vec, const float* __restrict__ len_,
                           const int* __restrict__ snd, const int* __restrict__ rcv,
                           float* __restrict__ g_pos, int E)
{
    int e = blockIdx.x * blockDim.x + threadIdx.x;
    if (e >= E) return;
    const float s3 = 1.7320508075688772f, s5 = 2.2360679774997896f, s15 = 3.8729833462074170f;
    float vx = vec[e*3], vy = vec[e*3+1], vz = vec[e*3+2];
    float len = len_[e];
    float inv = 1.0f/len;
    float x = vx*inv, y = vy*inv, z = vz*inv;
    const float* gs = g_sh + (size_t)e*9;
    float g1=gs[1], g2=gs[2], g3=gs[3], g4=gs[4], g5=gs[5], g6=gs[6], g7=gs[7], g8=gs[8];
    float gux = s3*g1 + s15*(y*g4 + z*g7 + x*g8);
    float guy = s3*g2 + s15*(x*g4 + z*g5 - y*g8);
    float guz = s3*g3 + s15*(y*g5 + x*g7) + 3.0f*s5*z*g6;
    float gdot = gux*x + guy*y + guz*z;
    float u = len * (1.0f/R_MAX_F);
    float fc = 0.f, dfc = 0.f;
    if (u < 1.0f) {
        float u2=u*u, u4=u2*u2, u5=u4*u, u6=u5*u;
        fc  = 1.0f - 21.0f*u5 + 35.0f*u6 - 15.0f*u6*u;
        dfc = (-105.0f*u4 + 210.0f*u5 - 105.0f*u6) * (1.0f/R_MAX_F);
    }
    float A = sqrtf(2.0f/R_MAX_F);
    float rinv = 1.0f/(len + 1e-9f);
    const float PI_F = 3.14159265358979f;
    float gr = 0.f;
    const float* ge = g_ef + (size_t)e*8;
#pragma unroll
    for (int b = 0; b < 8; ++b) {
        float np = (float)(b+1)*PI_F;
        float sarg = np*u;
        float s_ = sinf(sarg), c_ = cosf(sarg);
        float bess  = A*s_*rinv;
        float dbess = A*(c_*np*(1.0f/R_MAX_F)*rinv - s_*rinv*rinv);
        gr += ge[b]*(dbess*fc + bess*dfc);
    }
    float gvx = (gux - gdot*x)*inv + gr*x;
    float gvy = (guy - gdot*y)*inv + gr*y;
    float gvz = (guz - gdot*z)*inv + gr*z;
    int ss = snd[e], rr = rcv[e];
    atomicAdd(&g_pos[ss*3+0],  gvx); atomicAdd(&g_pos[ss*3+1],  gvy); atomicAdd(&g_pos[ss*3+2],  gvz);
    atomicAdd(&g_pos[rr*3+0], -gvx); atomicAdd(&g_pos[rr*3+1], -gvy); atomicAdd(&g_pos[rr*3+2], -gvz);
}

__global__ void forces_k(const float* __restrict__ g_pos, float* __restrict__ f, int N3)
{
    int i = blockIdx.x * blockDim.x + threadIdx.x;
    if (i >= N3) return;
    f[i] = -g_pos[i];
}

// ------------------------------- host driver -------------------------------

static inline int cdivi(long a, long b) { return (int)((a + b - 1) / b); }

extern "C" void kernel_launch(void* const* d_in, const int* in_sizes, int n_in,
                              void* d_out, int out_size, void* d_ws, size_t ws_size,
                              hipStream_t stream)
{
    const float* pos     = (const float*)d_in[0];
    const float* attrs   = (const float*)d_in[1];
    const float* charges = (const float*)d_in[2];
    const float* shifts  = (const float*)d_in[3];
    const float* W_emb   = (const float*)d_in[4];
    const float* ae      = (const float*)d_in[5];
    const float* W_r1    = (const float*)d_in[6];
    const float* W_r2    = (const float*)d_in[7];
    const float* W_mix   = (const float*)d_in[8];
    const float* W_sc    = (const float*)d_in[9];
    const float* w_poly  = (const float*)d_in[10];
    const float* W_prod  = (const float*)d_in[11];
    const float* w_e     = (const float*)d_in[12];
    const float* w_d     = (const float*)d_in[13];
    const int*   ei      = (const int*)d_in[14];
    const int*   batch   = (const int*)d_in[15];

    const int N = in_sizes[0] / 3;
    const int E = in_sizes[3] / 3;
    const int Z = in_sizes[1] / N;
    const int G = (out_size - 6*N) / 7;
    const int* snd = ei;
    const int* rcv = ei + E;

    float* out       = (float*)d_out;
    float* o_te      = out;                    // [G]
    float* o_contrib = out + G;                // [G,3]
    float* o_forces  = out + 4*G;              // [N,3]
    float* o_td      = out + 4*G + 3*N;        // [G,3]
    float* o_ad      = out + 7*G + 3*N;        // [N,3]

    float* ws = (float*)d_ws;
    size_t off = 0;
    auto take = [&](size_t n) { float* p = ws + off; off += n; return p; };
    float* b_vec      = take(3*(size_t)E);
    float* b_len      = take((size_t)E);
    float* b_sh       = take(9*(size_t)E);
    float* b_ef       = take(8*(size_t)E);
    float* b_h1pre[2] = { take(64*(size_t)E), take(64*(size_t)E) };
    float* b_w[2]     = { take(64*(size_t)E), take(64*(size_t)E) };
    float* b_nf0[2]   = { take(64*(size_t)N), take(64*(size_t)N) };
    float* b_msg[2]   = { take(576*(size_t)N), take(576*(size_t)N) };
    float* b_nfA      = take(576*(size_t)N);
    float* b_nfB      = take(576*(size_t)N);
    float* b_tmp9     = take(576*(size_t)N);   // agg / mp (fwd), g_agg (bwd)
    float* b_gnf      = take(576*(size_t)N);
    float* b_gA       = take(576*(size_t)N);
    float* b_gC       = take(576*(size_t)N);
    float* b_gw       = take(64*(size_t)E);
    float* b_gh       = take(64*(size_t)E);
    float* b_gef      = take(8*(size_t)E);
    float* b_gsh      = take(9*(size_t)E);
    float* b_gpos     = take(3*(size_t)N);
    (void)ws_size; (void)n_in;

    const int TPB = 256;
    hipMemsetAsync(d_out, 0, (size_t)out_size * 4, stream);
    hipMemsetAsync(b_gef, 0, 8*(size_t)E*4, stream);
    hipMemsetAsync(b_gsh, 0, 9*(size_t)E*4, stream);
    hipMemsetAsync(b_gpos, 0, 3*(size_t)N*4, stream);
    hipMemsetAsync(b_gnf, 0, 576*(size_t)N*4, stream);

    edge_geom_k<<<cdivi(E, TPB), TPB, 0, stream>>>(pos, shifts, snd, rcv, b_vec, b_len, b_sh, b_ef, E);
    node_init_k<<<cdivi((long)N*64, TPB), TPB, 0, stream>>>(attrs, W_emb, ae, batch, b_nfA, o_contrib, N, Z);

    const dim3 gE(cdivi(cdivi(E, 16), 4), 1);
    const dim3 gN9(cdivi(cdivi(N, 16), 4), 9);

    // ------------------------------ forward --------------------------------
    for (int t = 0; t < 2; ++t) {
        float* nf_in  = t ? b_nfB : b_nfA;
        float* nf_out = t ? b_nfA : b_nfB;
        // radial MLP: h1pre = ef @ W_r1 ; w = silu(h1pre) @ W_r2
        wmma_gemm<8,64,0,0,0,  8,1,0, 64,1, 64,1,0><<<gE, 128, 0, stream>>>(
            b_ef, W_r1 + (size_t)t*8*64, b_h1pre[t], E);
        wmma_gemm<64,64,0,0,1, 64,1,0, 64,1, 64,1,0><<<gE, 128, 0, stream>>>(
            b_h1pre[t], W_r2 + (size_t)t*64*64, b_w[t], E);
        save_ch0_k<<<cdivi((long)N*64, TPB), TPB, 0, stream>>>(nf_in, b_nf0[t], N);
        hipMemsetAsync(b_tmp9, 0, 576*(size_t)N*4, stream);
        edge_scatter_k<<<cdivi((long)E*64, TPB), TPB, 0, stream>>>(b_w[t], b_nf0[t], b_sh, snd, rcv, b_tmp9, E);
        // msg = per_l_linear(agg, W_mix)
        wmma_gemm<64,64,1,0,0, 576,9,1, 64,1, 576,9,1><<<gN9, 128, 0, stream>>>(
            b_tmp9, W_mix + (size_t)t*3*64*64, b_msg[t], N);
        // sc = per_l_linear(nf_in, W_sc) -> nf_out
        wmma_gemm<64,64,1,0,0, 576,9,1, 64,1, 576,9,1><<<gN9, 128, 0, stream>>>(
            nf_in, W_sc + (size_t)t*3*64*64, nf_out, N);
        node_poly_fwd_k<<<cdivi((long)N*64, TPB), TPB, 0, stream>>>(b_msg[t], w_poly + (size_t)t*64*3, b_tmp9, N);
        // nf_out += per_l_linear(msg*poly, W_prod)
        wmma_gemm<64,64,1,1,0, 576,9,1, 64,1, 576,9,1><<<gN9, 128, 0, stream>>>(
            b_tmp9, W_prod + (size_t)t*3*64*64, nf_out, N);
        energy_dipole_k<<<cdivi(N, TPB), TPB, 0, stream>>>(nf_out, w_e + (size_t)t*64,
            w_d + (size_t)t*64, batch, o_contrib, o_ad, N, t + 1);
    }
    finalize_dipole_k<<<cdivi(N, TPB), TPB, 0, stream>>>(o_ad, charges, pos, batch, o_td, N);

    // ------------------------------ backward -------------------------------
    for (int t = 1; t >= 0; --t) {
        add_we_k<<<cdivi((long)N*64, TPB), TPB, 0, stream>>>(b_gnf, w_e + (size_t)t*64, N);
        // g_mp = per_l_linear_T(g_nf, W_prod)   (transpose via swapped W strides)
        wmma_gemm<64,64,1,0,0, 576,9,1, 1,64, 576,9,1><<<gN9, 128, 0, stream>>>(
            b_gnf, W_prod + (size_t)t*3*64*64, b_gA, N);
        node_poly_bwd_k<<<cdivi((long)N*64, TPB), TPB, 0, stream>>>(b_gA, b_msg[t], w_poly + (size_t)t*64*3, N);
        // g_agg = per_l_linear_T(g_msg, W_mix)
        wmma_gemm<64,64,1,0,0, 576,9,1, 1,64, 576,9,1><<<gN9, 128, 0, stream>>>(
            b_gA, W_mix + (size_t)t*3*64*64, b_tmp9, N);
        // g_nf_in = per_l_linear_T(g_nf, W_sc)
        wmma_gemm<64,64,1,0,0, 576,9,1, 1,64, 576,9,1><<<gN9, 128, 0, stream>>>(
            b_gnf, W_sc + (size_t)t*3*64*64, b_gC, N);
        edge_bwd_k<<<cdivi(E, TPB), TPB, 0, stream>>>(b_tmp9, b_w[t], b_nf0[t], b_sh, snd, rcv,
            b_gw, b_gsh, b_gC, E);
        // radial backward: gh = g_w @ W_r2^T ; g_pre = gh*silu'(h1pre) ; g_ef += g_pre @ W_r1^T
        wmma_gemm<64,64,0,0,0, 64,1,0, 1,64, 64,1,0><<<gE, 128, 0, stream>>>(
            b_gw, W_r2 + (size_t)t*64*64, b_gh, E);
        silu_bwd_k<<<cdivi((long)E*64, TPB), TPB, 0, stream>>>(b_gh, b_h1pre[t], (long)E*64);
        wmma_gemm<64,8,0,1,0, 64,1,0, 1,64, 8,1,0><<<gE, 128, 0, stream>>>(
            b_gh, W_r1 + (size_t)t*8*64, b_gef, E);
        float* tmp = b_gnf; b_gnf = b_gC; b_gC = tmp;
    }
    geom_bwd_k<<<cdivi(E, TPB), TPB, 0, stream>>>(b_gsh, b_gef, b_vec, b_len, snd, rcv, b_gpos, E);
    forces_k<<<cdivi((long)N*3, TPB), TPB, 0, stream>>>(b_gpos, o_forces, N*3);
    total_energy_k<<<1, 256, 0, stream>>>(o_contrib, o_te, G);
}